// SharedDepthwiseConv_31988916420665
// MI455X (gfx1250) — compile-verified
//
#include <hip/hip_runtime.h>
#include <stdint.h>

#define N_  8
#define C_  512
#define H_  128
#define W_  128
#define TH  64                 // output rows per block
#define LROWS (TH + 8)         // +4 halo top/bottom (max dilation 4)
#define LCOLS (W_ + 8)         // +4 zero apron left/right

__global__ __launch_bounds__(256)
void dwconv_gated_kernel(const float* __restrict__ x,
                         const float* __restrict__ gate,
                         const float* __restrict__ wgt,
                         const float* __restrict__ bias,
                         float* __restrict__ out)
{
    __shared__ float tile[LROWS][LCOLS];   // 72 x 136 fp32 = 38.25 KB

    const int tid  = threadIdx.x;
    const int lane = tid & 31;
    const int wv   = tid >> 5;             // wave id 0..7
    const int h0   = blockIdx.x * TH;
    const int ch   = blockIdx.y;
    const int n    = blockIdx.z;

    const float* plane = x + ((size_t)n * C_ + ch) * (size_t)(H_ * W_);

    // ---------------- stage input tile: async global -> LDS ----------------
    // Rows h0-4 .. h0+TH+3; each valid row = one global_load_async_to_lds_b128
    // per wave (16B per lane = 512B per row). Aprons & OOB rows zero-filled.
    for (int r = wv; r < LROWS; r += 8) {
        const int ih = h0 - 4 + r;
        if (lane == 0) *(float4*)&tile[r][0]      = make_float4(0.f, 0.f, 0.f, 0.f);
        if (lane == 1) *(float4*)&tile[r][W_ + 4] = make_float4(0.f, 0.f, 0.f, 0.f);
        if (ih >= 0 && ih < H_) {
            uint32_t lds_byte = (uint32_t)(uintptr_t)&tile[r][4 + 4 * lane];
            const float* gptr = plane + (size_t)ih * W_ + 4 * lane;
            asm volatile("global_load_async_to_lds_b128 %0, %1, off"
                         :: "v"(lds_byte), "v"(gptr) : "memory");
        } else {
            *(float4*)&tile[r][4 + 4 * lane] = make_float4(0.f, 0.f, 0.f, 0.f);
        }
    }
    asm volatile("s_wait_asynccnt 0" ::: "memory");  // ASYNCcnt==0: our LDS writes landed
    __syncthreads();

    // ---------------- per-(n,c) uniform coefficients (scalar loads) --------
    const float g0 = gate[n * 3 + 0];
    const float g1 = gate[n * 3 + 1];
    const float g2 = gate[n * 3 + 2];
    float cf[3][9];
    #pragma unroll
    for (int k = 0; k < 9; ++k) {
        const float wk = wgt[ch * 9 + k];
        cf[0][k] = g0 * wk;   // dilation 1
        cf[1][k] = g1 * wk;   // dilation 2
        cf[2][k] = g2 * wk;   // dilation 4
    }
    const float gsum  = g0 + g1 + g2;
    const float cfc   = gsum * wgt[ch * 9 + 4];  // shared center tap (kh=1,kw=1, all d)
    const float binit = gsum * bias[ch];

    // ---------------- compute: 8 rows x 4 cols per thread ------------------
    const int cg = tid & 31;       // col group: cols 4*cg .. 4*cg+3
    const int rg = tid >> 5;       // row group: rows 8*rg .. 8*rg+7
    const int w0 = cg * 4;
    const int r0 = rg * 8;

    float acc[8][4];
    #pragma unroll
    for (int j = 0; j < 8; ++j)
        #pragma unroll
        for (int k = 0; k < 4; ++k)
            acc[j][k] = binit;

    const int DIL[3] = {1, 2, 4};

    #pragma unroll
    for (int ro = 0; ro < 16; ++ro) {                 // input rows r0-4 .. r0+11
        const float* wrow = &tile[r0 + ro][w0];       // 16B-aligned window base
        const float4 a = *(const float4*)(wrow);
        const float4 b = *(const float4*)(wrow + 4);
        const float4 c = *(const float4*)(wrow + 8);
        const float win[12] = {a.x, a.y, a.z, a.w,
                               b.x, b.y, b.z, b.w,
                               c.x, c.y, c.z, c.w};
        #pragma unroll
        for (int j = 0; j < 8; ++j) {
            const int dr = ro - 4 - j;                // input_row - output_row
            if (dr == 0) {
                // kh = 1 for every dilation; center column tap is shared.
                #pragma unroll
                for (int k = 0; k < 4; ++k)
                    acc[j][k] = fmaf(cfc, win[4 + k], acc[j][k]);
                #pragma unroll
                for (int di = 0; di < 3; ++di) {
                    const int d = DIL[di];
                    #pragma unroll
                    for (int kw = 0; kw < 3; kw += 2) {   // kw = 0, 2
                        const float cfv = cf[di][3 + kw];
                        const int base = 4 + d * (kw - 1);
                        #pragma unroll
                        for (int k = 0; k < 4; ++k)
                            acc[j][k] = fmaf(cfv, win[base + k], acc[j][k]);
                    }
                }
            } else {
                #pragma unroll
                for (int di = 0; di < 3; ++di) {
                    const int d = DIL[di];
                    if (dr == -d || dr == d) {            // compile-time after unroll
                        const int kh = dr / d + 1;
                        #pragma unroll
                        for (int kw = 0; kw < 3; ++kw) {
                            const float cfv = cf[di][kh * 3 + kw];
                            const int base = 4 + d * (kw - 1);
                            #pragma unroll
                            for (int k = 0; k < 4; ++k)
                                acc[j][k] = fmaf(cfv, win[base + k], acc[j][k]);
                        }
                    }
                }
            }
        }
    }

    // ---------------- store 8x float4 ----------------
    float* oplane = out + ((size_t)n * C_ + ch) * (size_t)(H_ * W_);
    #pragma unroll
    for (int j = 0; j < 8; ++j) {
        const int h = h0 + r0 + j;
        *(float4*)(oplane + (size_t)h * W_ + w0) =
            make_float4(acc[j][0], acc[j][1], acc[j][2], acc[j][3]);
    }
}

extern "C" void kernel_launch(void* const* d_in, const int* in_sizes, int n_in,
                              void* d_out, int out_size, void* d_ws, size_t ws_size,
                              hipStream_t stream) {
    (void)in_sizes; (void)n_in; (void)out_size; (void)d_ws; (void)ws_size;
    const float* x    = (const float*)d_in[0];
    const float* gate = (const float*)d_in[1];
    const float* wgt  = (const float*)d_in[2];
    const float* bias = (const float*)d_in[3];
    float* out = (float*)d_out;

    dim3 grid(H_ / TH, C_, N_);   // 2 x 512 x 8 = 8192 blocks, 256 threads (8 waves)
    dwconv_gated_kernel<<<grid, 256, 0, stream>>>(x, gate, wgt, bias, out);
}